// PseudoImageScatter_4243427688683
// MI455X (gfx1250) — compile-verified
//
#include <hip/hip_runtime.h>
#include <stdint.h>

// Problem constants (from the reference)
#define BB 4
#define PP 12000
#define CC 64
#define HH 512
#define WW 512

// -------------------------------------------------------------------------
// Kernel 1: zero the 256 MB canvas with b128 stores (harness poisons d_out,
// so every call must rewrite it; this pass is the bandwidth floor anyway).
// -------------------------------------------------------------------------
__global__ __launch_bounds__(256) void fill_zero_f4(float4* __restrict__ out, int n4) {
    int i = blockIdx.x * blockDim.x + threadIdx.x;
    const int stride = gridDim.x * blockDim.x;
    const float4 z = make_float4(0.f, 0.f, 0.f, 0.f);
    for (; i < n4; i += stride) {
        out[i] = z;                       // -> global_store_b128
    }
}

// -------------------------------------------------------------------------
// CDNA5 async data-mover helpers (inline asm: portable across toolchains).
// -------------------------------------------------------------------------

// Low 32 bits of a generic pointer to a __shared__ object == LDS byte offset
// (flat shared aperture keeps the workgroup-relative offset in addr[31:0]).
__device__ __forceinline__ unsigned lds_off_of(const void* p) {
    return (unsigned)(uintptr_t)p;
}

// All 32 lanes copy 8 bytes each: one full 256-byte pillar row -> LDS,
// asynchronously (tracked by ASYNCcnt, not LOADcnt).
__device__ __forceinline__ void async_copy_row(unsigned lds_dst,
                                               const float* __restrict__ grow,
                                               int lane) {
    const char* g = (const char*)grow + lane * 8;
    unsigned l = lds_dst + (unsigned)(lane * 8);
    asm volatile("global_load_async_to_lds_b64 %0, %1, off"
                 :: "v"(l), "v"(g)
                 : "memory");
}

// -------------------------------------------------------------------------
// Kernel 2: one wave per pillar, grid-stride. Double-buffered async staging:
// while the wave scatters row i from LDS, the TDM-side async path is already
// streaming row i+stride into the other buffer.
// -------------------------------------------------------------------------
__global__ __launch_bounds__(256) void pillar_scatter(
    const float* __restrict__ pf,      // [B, P, C] f32
    const int*   __restrict__ coords,  // [B, P, 4] i32 (b, y, x, z)
    float*       __restrict__ out)     // [B, C, H, W] f32 (already zeroed)
{
    // 8 waves per block, 2 buffers of 64 floats each per wave = 4 KB LDS.
    __shared__ float smem[8 * 2 * CC];

    const int lane      = threadIdx.x & 31;
    const int waveInBlk = threadIdx.x >> 5;
    const int wavesPerBlk = blockDim.x >> 5;
    const int nWaves    = gridDim.x * wavesPerBlk;
    const int total     = BB * PP;

    float* myRows = &smem[waveInBlk * 2 * CC];
    const unsigned myLds = lds_off_of(myRows);
    const unsigned bufBytes = CC * sizeof(float);    // 256 B per buffer

    int i = blockIdx.x * wavesPerBlk + waveInBlk;
    int cur = 0;

    if (i < total) {
        async_copy_row(myLds, pf + (size_t)i * CC, lane);   // prefetch first row
    }

    for (; i < total; i += nWaves) {
        const int nxt = i + nWaves;
        if (nxt < total) {
            // issue prefetch of next row into the other buffer, then wait
            // only for the OLDER load (done signals are in-order): cnt <= 1.
            async_copy_row(myLds + (cur ^ 1) * bufBytes, pf + (size_t)nxt * CC, lane);
            asm volatile("s_wait_asynccnt 1" ::: "memory");
        } else {
            asm volatile("s_wait_asynccnt 0" ::: "memory");
        }

        // coords[i] = (b, y, x, z) as one int4 (single b128 broadcast load)
        const int4 cc = ((const int4*)coords)[i];
        const int y = cc.y;
        const int x = cc.z;

        if ((unsigned)x < WW && (unsigned)y < HH) {
            const float* row = myRows + cur * CC;
            const int b = i / PP;
            const float v0 = row[lane];          // ds_load_b32
            const float v1 = row[lane + 32];     // ds_load_b32
            // out[b][c][y][x], channel planes are H*W apart
            const size_t base = (((size_t)(b * CC + lane) * HH) + (size_t)y) * WW + (size_t)x;
            out[base] = v0;
            out[base + (size_t)32 * HH * WW] = v1;
        }
        cur ^= 1;
    }
}

// -------------------------------------------------------------------------
// Launch: fill (256 MB write-limited) then scatter (12 MB payload).
// -------------------------------------------------------------------------
extern "C" void kernel_launch(void* const* d_in, const int* in_sizes, int n_in,
                              void* d_out, int out_size, void* d_ws, size_t ws_size,
                              hipStream_t stream) {
    (void)in_sizes; (void)n_in; (void)d_ws; (void)ws_size;

    const float* pf     = (const float*)d_in[0];   // [4,12000,64] f32
    const int*   coords = (const int*)d_in[1];     // [4,12000,4]  i32
    float*       out    = (float*)d_out;           // [4,64,512,512] f32

    const int n4 = out_size / 4;                   // 16,777,216 float4s
    fill_zero_f4<<<8192, 256, 0, stream>>>((float4*)out, n4);

    // 750 blocks * 8 waves = 6000 waves, 8 pillars per wave -> the double
    // buffer amortizes and there are plenty of waves to cover store latency.
    pillar_scatter<<<750, 256, 0, stream>>>(pf, coords, out);
}